// network_17927193494070
// MI455X (gfx1250) — compile-verified
//
#include <hip/hip_runtime.h>

typedef __attribute__((ext_vector_type(2))) float v2f;
typedef __attribute__((ext_vector_type(8))) float v8f;

#define B_ 8
#define C_ 64
#define H_ 512
#define W_ 512
#define HW_ (H_ * W_)
#define TW 64
#define TH 16

// ---------------------------------------------------------------------------
// Kernel 1: per-(b,c) plane sum. One block per plane, float4 streaming loads.
// ---------------------------------------------------------------------------
__global__ void sekg_plane_sum(const float* __restrict__ x, float* __restrict__ featsum) {
    const int plane = blockIdx.x;                 // 0..511
    const int tid = threadIdx.x;                  // 0..255
    const float4* xv = (const float4*)(x + (size_t)plane * HW_);
    float s = 0.f;
    // HW_/4 = 65536 float4 per plane
    for (int i = tid; i < HW_ / 4; i += 256) {
        float4 v = xv[i];
        s += v.x + v.y + v.z + v.w;
    }
    __shared__ float red[256];
    red[tid] = s;
    __syncthreads();
    for (int off = 128; off > 0; off >>= 1) {
        if (tid < off) red[tid] += red[tid + off];
        __syncthreads();
    }
    if (tid == 0) featsum[plane] = red[0];
}

// ---------------------------------------------------------------------------
// Kernel 2: SEKG head, single block of 128 threads (4 waves).
//   h1 = leakyrelu(feat @ w1^T + b1)      -> GEMM1: 4 waves x WMMA f32 16x16x4
//   k  = h1 @ w2^T + b2, normalized       -> GEMM2: wave 0 x WMMA f32 16x16x4
// A layout (16x4 f32): lane%16 = M, v0/v1 = K + (lane<16 ? 0 : 2)
// B layout (4x16 f32): lane%16 = N, v0/v1 = K + (lane<16 ? 0 : 2)
// C/D layout: lane%16 = N, vgpr i = M=i (lanes 0-15) / M=8+i (lanes 16-31)
// ---------------------------------------------------------------------------
__global__ void sekg_head(const float* __restrict__ featsum,
                          const float* __restrict__ w1, const float* __restrict__ b1,
                          const float* __restrict__ w2, const float* __restrict__ b2,
                          float* __restrict__ kout) {
    __shared__ float sA[16][64];   // feat, rows 8..15 zero
    __shared__ float sH[16][64];   // h1
    __shared__ float sK[8][9];
    __shared__ float sSum[8];

    const int tid = threadIdx.x;           // 0..127
    const int wv = tid >> 5;               // wave id 0..3
    const int lane = tid & 31;
    const int mrow = lane & 15;
    const int khi = (lane >> 4) << 1;      // 0 for lanes 0-15, 2 for lanes 16-31

    // Stage A = feat (mean = sum / HW), padded to 16 rows with zeros.
    const float inv_hw = 1.0f / (float)HW_;
    for (int idx = tid; idx < 16 * 64; idx += 128) {
        int m = idx >> 6, k = idx & 63;
        sA[m][k] = (m < B_) ? featsum[m * 64 + k] * inv_hw : 0.f;
    }
    __syncthreads();

    // GEMM1: each wave computes a 16-wide column tile of h1 = A @ w1^T.
    {
        const int ncol = wv * 16 + mrow;   // output channel 0..63
        v8f acc = {};
        for (int kk = 0; kk < 64; kk += 4) {
            v2f a, bb;
            a.x = sA[mrow][kk + khi];
            a.y = sA[mrow][kk + khi + 1];
            bb.x = w1[ncol * 64 + kk + khi];       // B[k][n] = w1[n][k]
            bb.y = w1[ncol * 64 + kk + khi + 1];
            acc = __builtin_amdgcn_wmma_f32_16x16x4_f32(
                false, a, false, bb, (short)0, acc, false, false);
        }
        const float bias = b1[ncol];
        const int mbase = (lane >> 4) << 3; // 0 or 8
        #pragma unroll
        for (int i = 0; i < 8; i++) {
            float v = acc[i] + bias;
            sH[mbase + i][ncol] = (v > 0.f) ? v : 0.2f * v;   // LeakyReLU(0.2)
        }
    }
    __syncthreads();

    // GEMM2: wave 0 computes k = h1 @ w2^T (N = 9 of 16 used).
    if (wv == 0) {
        const int j = mrow;                // 0..15; valid j < 9
        v8f acc = {};
        for (int kk = 0; kk < 64; kk += 4) {
            v2f a, bb;
            a.x = sH[mrow][kk + khi];
            a.y = sH[mrow][kk + khi + 1];
            bb.x = (j < 9) ? w2[j * 64 + kk + khi]     : 0.f;  // B[k][j] = w2[j][k]
            bb.y = (j < 9) ? w2[j * 64 + kk + khi + 1] : 0.f;
            acc = __builtin_amdgcn_wmma_f32_16x16x4_f32(
                false, a, false, bb, (short)0, acc, false, false);
        }
        if (lane < 16 && j < 9) {          // lanes 0-15 hold D rows 0..7
            const float bias = b2[j];
            #pragma unroll
            for (int i = 0; i < 8; i++) sK[i][j] = acc[i] + bias;
        }
    }
    __syncthreads();

    if (tid < 8) {
        float s = 0.f;
        for (int j = 0; j < 9; j++) s += sK[tid][j];
        sSum[tid] = s + 1e-6f;
    }
    __syncthreads();

    if (tid < B_ * 9) {
        int b = tid / 9, j = tid % 9;
        kout[tid] = sK[b][j] / sSum[b];
    }
}

// ---------------------------------------------------------------------------
// Kernel 3: depthwise 3x3 conv, per-sample kernel, zero padding.
// One block per (plane, 64x16 tile). LDS halo tile 18x66 -> 1.16x read amp.
// ---------------------------------------------------------------------------
__global__ void sekg_conv(const float* __restrict__ x, const float* __restrict__ kbuf,
                          float* __restrict__ out) {
    __shared__ float tile[TH + 2][TW + 2];

    const int plane = blockIdx.z;                   // 0..511 = b*64 + c
    const int b = plane >> 6;
    const int x0 = blockIdx.x * TW;
    const int y0 = blockIdx.y * TH;
    const int tid = threadIdx.x;                    // 0..255

    const float* kp = kbuf + b * 9;
    const float k0 = kp[0], k1 = kp[1], k2 = kp[2];
    const float k3 = kp[3], k4 = kp[4], k5 = kp[5];
    const float k6 = kp[6], k7 = kp[7], k8 = kp[8];

    const float* xp = x + (size_t)plane * HW_;

    for (int idx = tid; idx < (TH + 2) * (TW + 2); idx += 256) {
        int ly = idx / (TW + 2);
        int lx = idx - ly * (TW + 2);
        int gy = y0 + ly - 1;
        int gx = x0 + lx - 1;
        float v = 0.f;
        if (gy >= 0 && gy < H_ && gx >= 0 && gx < W_) v = xp[gy * W_ + gx];
        tile[ly][lx] = v;
    }
    __syncthreads();

    const int cx = tid & 63;       // column within tile
    const int ry = tid >> 6;       // 0..3
    float* op = out + (size_t)plane * HW_;
    #pragma unroll
    for (int rr = 0; rr < 4; rr++) {
        const int r = ry + rr * 4; // row within tile, 0..15
        float s = k0 * tile[r    ][cx] + k1 * tile[r    ][cx + 1] + k2 * tile[r    ][cx + 2]
                + k3 * tile[r + 1][cx] + k4 * tile[r + 1][cx + 1] + k5 * tile[r + 1][cx + 2]
                + k6 * tile[r + 2][cx] + k7 * tile[r + 2][cx + 1] + k8 * tile[r + 2][cx + 2];
        op[(size_t)(y0 + r) * W_ + x0 + cx] = s;
    }
}

// ---------------------------------------------------------------------------
extern "C" void kernel_launch(void* const* d_in, const int* in_sizes, int n_in,
                              void* d_out, int out_size, void* d_ws, size_t ws_size,
                              hipStream_t stream) {
    const float* x  = (const float*)d_in[0];
    const float* w1 = (const float*)d_in[1];
    const float* b1 = (const float*)d_in[2];
    const float* w2 = (const float*)d_in[3];
    const float* b2 = (const float*)d_in[4];
    float* out = (float*)d_out;

    float* ws   = (float*)d_ws;
    float* feat = ws;          // 512 floats: per-plane sums
    float* kbuf = ws + 512;    // 72 floats: normalized per-sample 3x3 kernels

    sekg_plane_sum<<<B_ * C_, 256, 0, stream>>>(x, feat);
    sekg_head<<<1, 128, 0, stream>>>(feat, w1, b1, w2, b2, kbuf);
    sekg_conv<<<dim3(W_ / TW, H_ / TH, B_ * C_), 256, 0, stream>>>(x, kbuf, out);
}